// GatedAttentionAggregator_24893630447804
// MI455X (gfx1250) — compile-verified
//
#include <hip/hip_runtime.h>
#include <hip/hip_bf16.h>

// ---------------------------------------------------------------------------
// GaAN-style gated attention aggregator for MI455X (gfx1250).
// Dense transforms use V_WMMA_F32_16X16X4_F32 (full fp32 precision).
// Edge scatters use float atomics; batch-norm/gate fused in a finalize pass.
// ---------------------------------------------------------------------------

typedef __attribute__((ext_vector_type(2))) float v2f;
typedef __attribute__((ext_vector_type(8))) float v8f;

#define DIN 128

__device__ __forceinline__ float wave_sum32(float v) {
#pragma unroll
    for (int off = 16; off > 0; off >>= 1) v += __shfl_xor(v, off, 32);
    return v;
}

// Order-independent float atomic max (works for mixed signs, -inf init).
__device__ __forceinline__ void atomicMaxF(float* addr, float val) {
    if (val >= 0.0f) {
        atomicMax(reinterpret_cast<int*>(addr), __float_as_int(val));
    } else {
        atomicMin(reinterpret_cast<unsigned int*>(addr), __float_as_uint(val));
    }
}

// ---------------------------------------------------------------------------
// WMMA GEMM: Out[N, NT*16] = act(A[N,128] @ W[128, NT*16] + bias)
// One wave computes a 16-row strip across all NT column tiles.
// A fragment (16x4 f32): lane holds A[m][2*kh + {0,1}], m=lane&15, kh=lane>>4
// B fragment (4x16 f32): lane holds W[2*kh + {0,1}][n],  n=lane&15
// C tile (16x16 f32):    vgpr j <-> row j + 8*kh, col n
// ---------------------------------------------------------------------------
template <int NT, bool RELU>
__global__ __launch_bounds__(256) void gemm_wmma_kernel(
    const float* __restrict__ A, const float* __restrict__ W,
    const float* __restrict__ bias, float* __restrict__ Out, int N) {
    constexpr int DOUT = NT * 16;
    const int wave = threadIdx.x >> 5;
    const int lane = threadIdx.x & 31;
    const int tile = blockIdx.x * 8 + wave;     // 16-node row tile
    if (tile * 16 >= N) return;                  // wave-uniform: EXEC stays full
    const int node0 = tile * 16;
    const int m  = lane & 15;
    const int kh = lane >> 4;

    const float* arow = A + (size_t)(node0 + m) * DIN + 2 * kh; // + k0 later
    const float* wcol = W + (size_t)(2 * kh) * DOUT + m;        // + k0*DOUT + t*16

    v8f zero = {};
    v8f acc[NT];
#pragma unroll
    for (int t = 0; t < NT; ++t) acc[t] = zero;

#pragma unroll 4
    for (int k0 = 0; k0 < DIN; k0 += 4) {
        float2 af = *reinterpret_cast<const float2*>(arow + k0);
        v2f a; a.x = af.x; a.y = af.y;
        const float* wk = wcol + (size_t)k0 * DOUT;
#pragma unroll
        for (int t = 0; t < NT; ++t) {
            v2f b;
            b.x = wk[t * 16];
            b.y = wk[t * 16 + DOUT];
            acc[t] = __builtin_amdgcn_wmma_f32_16x16x4_f32(
                /*neg_a=*/false, a, /*neg_b=*/false, b,
                /*c_mod=*/(short)0, acc[t], /*reuse_a=*/false, /*reuse_b=*/false);
        }
    }

#pragma unroll
    for (int t = 0; t < NT; ++t) {
        const int col = t * 16 + m;
        const float bv = bias ? bias[col] : 0.0f;
#pragma unroll
        for (int j = 0; j < 8; ++j) {
            const int row = node0 + j + 8 * kh;
            float v = acc[t][j] + bv;
            if (RELU) v = v > 0.0f ? v : 0.0f;
            Out[(size_t)row * DOUT + col] = v;
        }
    }
}

// ---------------------------------------------------------------------------
// Attention scores: a_self = lrelu(h0 @ att[:128]), a_neigh = lrelu(h1 @ att[128:])
// ---------------------------------------------------------------------------
__global__ __launch_bounds__(256) void attn_kernel(
    const float* __restrict__ h0, const float* __restrict__ h1,
    const float* __restrict__ att, float* __restrict__ a_self,
    float* __restrict__ a_neigh, int N) {
    const int wave = threadIdx.x >> 5;
    const int lane = threadIdx.x & 31;
    const int i = blockIdx.x * 8 + wave;
    if (i >= N) return;
    float s0 = 0.0f, s1 = 0.0f;
#pragma unroll
    for (int t = 0; t < 4; ++t) {
        const int d = lane + 32 * t;
        s0 += h0[(size_t)i * DIN + d] * att[d];
        s1 += h1[(size_t)i * DIN + d] * att[DIN + d];
    }
    s0 = wave_sum32(s0);
    s1 = wave_sum32(s1);
    if (lane == 0) {
        a_self[i]  = s0 > 0.0f ? s0 : 0.2f * s0;
        a_neigh[i] = s1 > 0.0f ? s1 : 0.2f * s1;
    }
}

// ---------------------------------------------------------------------------
// Per-call workspace init (workspace is poisoned / stale between calls).
// ---------------------------------------------------------------------------
__global__ __launch_bounds__(256) void init_kernel(
    float* __restrict__ neigh_zj, float* __restrict__ neigh_mean,
    float* __restrict__ agg, int N) {
    const int idx = blockIdx.x * blockDim.x + threadIdx.x;
    if (idx < N * 64) neigh_zj[idx] = -__builtin_inff();
    if (idx < N * DIN) {
        neigh_mean[idx] = 0.0f;
        agg[idx] = 0.0f;
    }
}

// ---------------------------------------------------------------------------
// Fused edge scatter: segment_max(zj), segment_sum(vals*feat), segment_sum(e*h1)
// One wave per edge.
// ---------------------------------------------------------------------------
__global__ __launch_bounds__(256) void edge_kernel(
    const int* __restrict__ rows, const int* __restrict__ cols,
    const float* __restrict__ vals, const float* __restrict__ feat,
    const float* __restrict__ zj, const float* __restrict__ h1,
    const float* __restrict__ a_self, const float* __restrict__ a_neigh,
    float* __restrict__ neigh_zj, float* __restrict__ neigh_mean,
    float* __restrict__ agg, int E) {
    const int wave = threadIdx.x >> 5;
    const int lane = threadIdx.x & 31;
    const int e = blockIdx.x * 8 + wave;
    if (e >= E) return;
    const int r = rows[e];
    const int c = cols[e];
    const float v = vals[e];
    const float es = a_self[r] + a_neigh[c];

#pragma unroll
    for (int t = 0; t < 2; ++t) {
        const int d = lane + 32 * t;
        atomicMaxF(&neigh_zj[(size_t)r * 64 + d], zj[(size_t)c * 64 + d]);
    }
#pragma unroll
    for (int t = 0; t < 4; ++t) {
        const int d = lane + 32 * t;
        atomicAdd(&neigh_mean[(size_t)r * DIN + d], v * feat[(size_t)c * DIN + d]);
        atomicAdd(&agg[(size_t)r * DIN + d], es * h1[(size_t)c * DIN + d]);
    }
}

// ---------------------------------------------------------------------------
// Finalize: gate = [feat, nz(neigh_zj), neigh_mean] @ W_gate ; agg *= gate ;
// out = bn(h0; s0,o0) + bn(agg; s1,o1)   (per-node mean/var over 128 features)
// One wave per node.
// ---------------------------------------------------------------------------
__global__ __launch_bounds__(256) void finalize_kernel(
    const float* __restrict__ feat, const float* __restrict__ neigh_zj,
    const float* __restrict__ neigh_mean, const float* __restrict__ Wg,
    const float* __restrict__ h0, const float* __restrict__ agg,
    const float* __restrict__ scale0, const float* __restrict__ offset0,
    const float* __restrict__ scale1, const float* __restrict__ offset1,
    float* __restrict__ out, int N) {
    const int wave = threadIdx.x >> 5;
    const int lane = threadIdx.x & 31;
    const int i = blockIdx.x * 8 + wave;
    if (i >= N) return;

    // gate: dot([feat | fixed(neigh_zj) | neigh_mean], W_gate[320])
    float g = 0.0f;
#pragma unroll
    for (int t = 0; t < 4; ++t) {
        const int d = lane + 32 * t;
        g += feat[(size_t)i * DIN + d] * Wg[d];
        g += neigh_mean[(size_t)i * DIN + d] * Wg[192 + d];
    }
#pragma unroll
    for (int t = 0; t < 2; ++t) {
        const int d = lane + 32 * t;
        float z = neigh_zj[(size_t)i * 64 + d];
        z = (z > -__builtin_inff()) ? z : 0.0f;   // empty rows -> 0
        g += z * Wg[128 + d];
    }
    g = wave_sum32(g);   // broadcast to all lanes via xor-reduction

    float x0[4], xa[4];
    float s0 = 0.0f, q0 = 0.0f, s1 = 0.0f, q1 = 0.0f;
#pragma unroll
    for (int t = 0; t < 4; ++t) {
        const int d = lane + 32 * t;
        x0[t] = h0[(size_t)i * DIN + d];
        xa[t] = agg[(size_t)i * DIN + d] * g;
        s0 += x0[t]; q0 += x0[t] * x0[t];
        s1 += xa[t]; q1 += xa[t] * xa[t];
    }
    s0 = wave_sum32(s0) * (1.0f / 128.0f);
    q0 = wave_sum32(q0) * (1.0f / 128.0f);
    s1 = wave_sum32(s1) * (1.0f / 128.0f);
    q1 = wave_sum32(q1) * (1.0f / 128.0f);
    const float r0 = rsqrtf(q0 - s0 * s0 + 1e-9f);
    const float r1 = rsqrtf(q1 - s1 * s1 + 1e-9f);

#pragma unroll
    for (int t = 0; t < 4; ++t) {
        const int d = lane + 32 * t;
        out[(size_t)i * DIN + d] =
            (x0[t] - s0) * scale0[d] * r0 + offset0[d] +
            (xa[t] - s1) * scale1[d] * r1 + offset1[d];
    }
}

// ---------------------------------------------------------------------------
// Host launcher
// ---------------------------------------------------------------------------
extern "C" void kernel_launch(void* const* d_in, const int* in_sizes, int n_in,
                              void* d_out, int out_size, void* d_ws, size_t ws_size,
                              hipStream_t stream) {
    const int*   rows    = (const int*)d_in[0];
    const int*   cols    = (const int*)d_in[1];
    const float* vals    = (const float*)d_in[2];
    const float* feat    = (const float*)d_in[3];
    const float* W0      = (const float*)d_in[4];
    const float* b0      = (const float*)d_in[5];
    const float* W1      = (const float*)d_in[6];
    const float* b1      = (const float*)d_in[7];
    const float* att     = (const float*)d_in[8];
    const float* W_pool  = (const float*)d_in[9];
    const float* W_gate  = (const float*)d_in[10];
    const float* offset0 = (const float*)d_in[11];
    const float* scale0  = (const float*)d_in[12];
    const float* offset1 = (const float*)d_in[13];
    const float* scale1  = (const float*)d_in[14];

    const int E = in_sizes[0];
    const int N = in_sizes[3] / DIN;

    float* ws = (float*)d_ws;
    float* h0         = ws;                 ws += (size_t)N * DIN;
    float* h1         = ws;                 ws += (size_t)N * DIN;
    float* zj         = ws;                 ws += (size_t)N * 64;
    float* a_self     = ws;                 ws += (size_t)N;
    float* a_neigh    = ws;                 ws += (size_t)N;
    float* neigh_zj   = ws;                 ws += (size_t)N * 64;
    float* neigh_mean = ws;                 ws += (size_t)N * DIN;
    float* agg        = ws;                 ws += (size_t)N * DIN;

    const int tiles = (N + 15) / 16;
    const int gemm_blocks = (tiles + 7) / 8;
    const int node_blocks = (N + 7) / 8;
    const int edge_blocks = (E + 7) / 8;
    const int init_blocks = (N * DIN + 255) / 256;

    init_kernel<<<init_blocks, 256, 0, stream>>>(neigh_zj, neigh_mean, agg, N);

    gemm_wmma_kernel<8, true ><<<gemm_blocks, 256, 0, stream>>>(feat, W0, b0, h0, N);
    gemm_wmma_kernel<8, true ><<<gemm_blocks, 256, 0, stream>>>(feat, W1, b1, h1, N);
    gemm_wmma_kernel<4, false><<<gemm_blocks, 256, 0, stream>>>(feat, W_pool, nullptr, zj, N);

    attn_kernel<<<node_blocks, 256, 0, stream>>>(h0, h1, att, a_self, a_neigh, N);

    edge_kernel<<<edge_blocks, 256, 0, stream>>>(rows, cols, vals, feat, zj, h1,
                                                 a_self, a_neigh,
                                                 neigh_zj, neigh_mean, agg, E);

    finalize_kernel<<<node_blocks, 256, 0, stream>>>(feat, neigh_zj, neigh_mean, W_gate,
                                                     h0, agg, scale0, offset0,
                                                     scale1, offset1,
                                                     (float*)d_out, N);
}